// GCN_FedTAD_6828998000936
// MI455X (gfx1250) — compile-verified
//
#include <hip/hip_runtime.h>
#include <hip/hip_bf16.h>
#include <math.h>

typedef float v2f __attribute__((ext_vector_type(2)));
typedef float v8f __attribute__((ext_vector_type(8)));

// ---------------------------------------------------------------------------
// deg[i] = 1 (self loop), then += 1 per incoming edge, then deg = rsqrt(deg)
// ---------------------------------------------------------------------------
__global__ void deg_init_kernel(float* __restrict__ deg, int n) {
    int i = blockIdx.x * blockDim.x + threadIdx.x;
    if (i < n) deg[i] = 1.0f;
}

__global__ void deg_accum_kernel(const long long* __restrict__ dst,
                                 float* __restrict__ deg, int E) {
    int i = blockIdx.x * blockDim.x + threadIdx.x;
    if (i < E) atomicAdd(&deg[(int)dst[i]], 1.0f);
}

__global__ void deg_rsqrt_kernel(float* __restrict__ deg, int n) {
    int i = blockIdx.x * blockDim.x + threadIdx.x;
    if (i < n) {
        float d = deg[i];
        deg[i] = (d > 0.0f) ? rsqrtf(d) : 0.0f;
    }
}

// ---------------------------------------------------------------------------
// Zero fill
// ---------------------------------------------------------------------------
__global__ void zero_f32_kernel(float* __restrict__ p, size_t n) {
    size_t i = (size_t)blockIdx.x * blockDim.x + threadIdx.x;
    if (i < n) p[i] = 0.0f;
}

// ---------------------------------------------------------------------------
// H[nrows,M] = X[nrows,K] @ W[K,M] via V_WMMA_F32_16X16X4_F32.
// One wave per 16x16 output tile; K stepped by 4, unrolled x8.
// Branch-free inner loop: row AND col indices are clamped for loads; the
// out-of-range lanes compute garbage that is masked only at the final store.
// A frag (16x4 f32): lane l<16 -> row l, K=k..k+1 ; lanes 16-31 -> K=k+2..k+3
// B frag (4x16 f32): mirrored K split across lane halves, N = lane&15
// C/D (8 VGPR): vgpr v -> row v (lanes 0-15) / row v+8 (lanes 16-31)
// ---------------------------------------------------------------------------
__global__ void gemm_wmma_f32_kernel(const float* __restrict__ X,
                                     const float* __restrict__ W,
                                     float* __restrict__ H,
                                     int nrows, int K, int M) {
    int waveId = (int)((blockIdx.x * (size_t)blockDim.x + threadIdx.x) >> 5);
    int lane   = threadIdx.x & 31;
    int half   = lane >> 4;      // 0: lanes 0-15, 1: lanes 16-31
    int l      = lane & 15;

    int ntilesN = (M + 15) >> 4;
    int ntilesM = (nrows + 15) >> 4;
    if (waveId >= ntilesM * ntilesN) return;   // wave-uniform: EXEC stays all-1

    int mt = waveId / ntilesN;
    int nt = waveId % ntilesN;

    int row  = mt * 16 + l;
    int rowc = row < nrows ? row : nrows - 1;  // clamp (dup loads are harmless)
    int col  = nt * 16 + l;
    int colc = col < M ? col : M - 1;          // clamp (masked at store)

    const float* xrow = X + (size_t)rowc * K + 2 * half;   // + k each step
    const float* wcol = W + colc + (size_t)(2 * half) * M; // + k*M each step

    v8f c = {};
#pragma unroll 8
    for (int k = 0; k < K; k += 4) {
        v2f a = *(const v2f*)(xrow + k);       // 8B aligned: K % 4 == 0
        v2f b;
        b.x = wcol[(size_t)k * M];
        b.y = wcol[(size_t)k * M + M];
        c = __builtin_amdgcn_wmma_f32_16x16x4_f32(
                /*neg_a=*/false, a, /*neg_b=*/false, b,
                /*c_mod=*/(short)0, c, /*reuse_a=*/false, /*reuse_b=*/false);
    }

    if (col < M) {
#pragma unroll
        for (int v = 0; v < 8; ++v) {
            int r = mt * 16 + v + 8 * half;
            if (r < nrows) H[(size_t)r * M + col] = c[v];
        }
    }
}

// ---------------------------------------------------------------------------
// Gather-scale-scatter: one wave per (edge | self-loop), lane-per-feature.
// agg[dst] += h[src] * dinv[src]*dinv[dst]. Edge indices / norm factor are
// wave-uniform -> forced to SGPRs via readfirstlane; VMEM does only the
// coalesced feature gather + L2-resident f32 atomics.
// ---------------------------------------------------------------------------
__global__ void scatter_edges_kernel(const float* __restrict__ h,
                                     const long long* __restrict__ src,
                                     const long long* __restrict__ dst,
                                     const float* __restrict__ dinv,
                                     float* __restrict__ agg,
                                     int E, int n, int F) {
    long long wid = (((long long)blockIdx.x * blockDim.x) + threadIdx.x) >> 5;
    int lane = threadIdx.x & 31;
    long long total = (long long)E + n;
    if (wid >= total) return;

    int s, d;
    if (wid < E) {
        s = __builtin_amdgcn_readfirstlane((int)src[wid]);
        d = __builtin_amdgcn_readfirstlane((int)dst[wid]);
    } else {
        s = d = __builtin_amdgcn_readfirstlane((int)(wid - E));
    }

    float w = __builtin_amdgcn_readfirstlane(dinv[s]) *
              __builtin_amdgcn_readfirstlane(dinv[d]);
    const float* hp = h   + (size_t)s * F;
    float*       ap = agg + (size_t)d * F;
    for (int f = lane; f < F; f += 32)
        atomicAdd(ap + f, hp[f] * w);
}

// ---------------------------------------------------------------------------
// In-place a[i] = relu(a[i] + bias[i % F])
// ---------------------------------------------------------------------------
__global__ void bias_relu_kernel(float* __restrict__ a,
                                 const float* __restrict__ bias,
                                 size_t n, int F) {
    size_t i = (size_t)blockIdx.x * blockDim.x + threadIdx.x;
    if (i < n) {
        float v = a[i] + bias[(int)(i % (size_t)F)];
        a[i] = fmaxf(v, 0.0f);
    }
}

// ---------------------------------------------------------------------------
// In-place per-node: out = log_softmax(out + b2). 40 logits, one wave32 per
// node; lane covers cols {lane, lane+32}; wave reductions via shfl_xor.
// ---------------------------------------------------------------------------
__global__ void bias_logsoftmax40_kernel(float* __restrict__ out,
                                         const float* __restrict__ bias,
                                         int n) {
    int wid  = (int)(((size_t)blockIdx.x * blockDim.x + threadIdx.x) >> 5);
    int lane = threadIdx.x & 31;
    if (wid >= n) return;

    float* row = out + (size_t)wid * 40;
    bool v1 = lane < 8;
    float a0 = row[lane] + bias[lane];
    float a1 = v1 ? (row[32 + lane] + bias[32 + lane]) : -3.402823466e38f;

    float m = fmaxf(a0, a1);
#pragma unroll
    for (int off = 16; off > 0; off >>= 1)
        m = fmaxf(m, __shfl_xor(m, off, 32));

    float e = expf(a0 - m) + (v1 ? expf(a1 - m) : 0.0f);
#pragma unroll
    for (int off = 16; off > 0; off >>= 1)
        e += __shfl_xor(e, off, 32);

    float ls = m + logf(e);
    row[lane] = a0 - ls;
    if (v1) row[32 + lane] = a1 - ls;
}

// ---------------------------------------------------------------------------
extern "C" void kernel_launch(void* const* d_in, const int* in_sizes, int n_in,
                              void* d_out, int out_size, void* d_ws, size_t ws_size,
                              hipStream_t stream) {
    const float*     x   = (const float*)d_in[0];
    const long long* ei  = (const long long*)d_in[1];   // int64 edge_index [2,E]
    const float*     W1  = (const float*)d_in[2];
    const float*     b1  = (const float*)d_in[3];
    const float*     W2  = (const float*)d_in[4];
    const float*     b2  = (const float*)d_in[5];
    float*           out = (float*)d_out;

    const int IN  = 256, HID = 128, OUT = 40;
    const int E   = in_sizes[1] / 2;
    const int n   = in_sizes[0] / IN;

    const long long* srcp = ei;       // row 0
    const long long* dstp = ei + E;   // row 1

    // Workspace layout (floats): deg[n] | h1[n*HID] | agg1[n*HID]
    float* deg  = (float*)d_ws;
    float* h1   = deg + n;
    float* agg1 = h1 + (size_t)n * HID;
    float* z2   = h1;                 // reuse: h1 dead once agg1 is built

    const int B = 256;                // 8 wave32 per block
    auto cdiv = [](long long a, long long b) { return (int)((a + b - 1) / b); };

    // degrees -> dinv
    deg_init_kernel <<<cdiv(n, B), B, 0, stream>>>(deg, n);
    deg_accum_kernel<<<cdiv(E, B), B, 0, stream>>>(dstp, deg, E);
    deg_rsqrt_kernel<<<cdiv(n, B), B, 0, stream>>>(deg, n);

    // Layer 1: h1 = x @ W1 (fp32 WMMA)
    {
        long long tiles = (long long)((n + 15) / 16) * ((HID + 15) / 16);
        gemm_wmma_f32_kernel<<<cdiv(tiles, 8), B, 0, stream>>>(x, W1, h1, n, IN, HID);
    }
    zero_f32_kernel<<<cdiv((long long)n * HID, B), B, 0, stream>>>(agg1, (size_t)n * HID);
    scatter_edges_kernel<<<cdiv(((long long)E + n) * 32, B), B, 0, stream>>>(
        h1, srcp, dstp, deg, agg1, E, n, HID);
    bias_relu_kernel<<<cdiv((long long)n * HID, B), B, 0, stream>>>(
        agg1, b1, (size_t)n * HID, HID);

    // Layer 2: z2 = relu_out @ W2 (fp32 WMMA, N padded to 48, masked store)
    {
        long long tiles = (long long)((n + 15) / 16) * ((OUT + 15) / 16);
        gemm_wmma_f32_kernel<<<cdiv(tiles, 8), B, 0, stream>>>(agg1, W2, z2, n, HID, OUT);
    }
    zero_f32_kernel<<<cdiv((long long)n * OUT, B), B, 0, stream>>>(out, (size_t)n * OUT);
    scatter_edges_kernel<<<cdiv(((long long)E + n) * 32, B), B, 0, stream>>>(
        z2, srcp, dstp, deg, out, E, n, OUT);

    // bias + log_softmax, in place on d_out
    bias_logsoftmax40_kernel<<<cdiv((long long)n * 32, B), B, 0, stream>>>(out, b2, n);
}